// MLPAttention_76845554860069
// MI455X (gfx1250) — compile-verified
//
#include <hip/hip_runtime.h>
#include <math.h>

typedef __attribute__((ext_vector_type(2))) float v2f;
typedef __attribute__((ext_vector_type(8))) float v8f;

#define NEG_INF (-1e6f)

// Problem constants (B=4, Lq=64, Lk=1024, D=256, U=256)
#define BB 4
#define LQ 64
#define LK 1024
#define DD 256
#define UU 256

// ---------------------------------------------------------------------------
// Kernel 1: u2[d] = sum_u Wq[d,u] * v[u]   (Wq is the matrix applied to `key`
// in the reference). One block of 256 threads; v staged through LDS.
// ---------------------------------------------------------------------------
__global__ __launch_bounds__(256) void mlpattn_prep(
    const float* __restrict__ Wq, const float* __restrict__ v,
    float* __restrict__ u2) {
  __shared__ __align__(16) float vs[UU];
  const int t = threadIdx.x;
  vs[t] = v[t];
  __syncthreads();
  const float* row = Wq + (size_t)t * UU;
  float acc = 0.f;
#pragma unroll 8
  for (int u = 0; u < UU; ++u) acc = fmaf(row[u], vs[u], acc);
  u2[t] = acc;
}

// ---------------------------------------------------------------------------
// Kernel 2: per batch b, sk[k] = key[b,k,:]·u2 ; masked softmax over k with
// the reference's -1e6 fill (masked exp underflows to exactly 0 in fp32).
// Writes w[b,k]. 4 blocks x 256 threads; 4 rows per thread, float4 streams.
// ---------------------------------------------------------------------------
__global__ __launch_bounds__(256) void mlpattn_softmax(
    const float* __restrict__ key, const int* __restrict__ valid_len,
    const float* __restrict__ u2, float* __restrict__ w) {
  __shared__ __align__(16) float us[UU];
  __shared__ float sk[LK];
  __shared__ float red[256];
  const int t = threadIdx.x;
  const int b = blockIdx.x;
  us[t] = u2[t];
  __syncthreads();

  const int vl = valid_len[b];
  const float* kb = key + (size_t)b * LK * DD;

  float lmax = NEG_INF;
#pragma unroll
  for (int rr = 0; rr < 4; ++rr) {
    const int r = t + rr * 256;
    const float4* row = (const float4*)(kb + (size_t)r * DD);
    const float4* uv4 = (const float4*)us;
    float acc = 0.f;
#pragma unroll 8
    for (int d4 = 0; d4 < DD / 4; ++d4) {
      float4 kv = row[d4];
      float4 uv = uv4[d4];
      acc = fmaf(kv.x, uv.x, acc);
      acc = fmaf(kv.y, uv.y, acc);
      acc = fmaf(kv.z, uv.z, acc);
      acc = fmaf(kv.w, uv.w, acc);
    }
    const float s = (r < vl) ? acc : NEG_INF;
    sk[r] = s;
    lmax = fmaxf(lmax, s);
  }
  // block max
  red[t] = lmax;
  __syncthreads();
  for (int s = 128; s > 0; s >>= 1) {
    if (t < s) red[t] = fmaxf(red[t], red[t + s]);
    __syncthreads();
  }
  const float m = red[0];
  __syncthreads();

  // exp + block sum (masked entries are exactly 0: expf(-1e6 - m) underflows)
  float lsum = 0.f;
#pragma unroll
  for (int rr = 0; rr < 4; ++rr) {
    const int r = t + rr * 256;
    const float e = (r < vl) ? __expf(sk[r] - m) : 0.f;
    sk[r] = e;
    lsum += e;
  }
  red[t] = lsum;
  __syncthreads();
  for (int s = 128; s > 0; s >>= 1) {
    if (t < s) red[t] = red[t] + red[t + s];
    __syncthreads();
  }
  const float inv = 1.f / red[0];
#pragma unroll
  for (int rr = 0; rr < 4; ++rr) {
    const int r = t + rr * 256;
    w[(size_t)b * LK + r] = sk[r] * inv;
  }
}

// ---------------------------------------------------------------------------
// Kernel 3: out[b,q,:] = w[b,:] @ value[b]  via V_WMMA_F32_16X16X4_F32.
// One wave32 per (batch, 16-wide N tile). A-matrix 16x4 f32 layout:
//   lanes 0-15: {K0,K1} in the 2 VGPRs, lanes 16-31: {K2,K3}; rows (M) are
//   all equal to w, so each lane just broadcasts w[k].
// B-matrix 4x16 f32: VGPR0/1 hold rows K0/K1 on lanes 0-15 and K2/K3 on
//   lanes 16-31, N = lane&15 -> two coalesced 64B loads per K-step.
// D rows are identical (A rows identical), so the tile is broadcast-stored
// to all 64 query rows.
// ---------------------------------------------------------------------------
__global__ __launch_bounds__(32) void mlpattn_av_wmma(
    const float* __restrict__ w, const float* __restrict__ value,
    float* __restrict__ out) {
  const int lane = threadIdx.x;
  const int half = lane >> 4;  // 0: K pair {0,1}, 1: K pair {2,3}
  const int l16 = lane & 15;   // N within tile
  const int b = blockIdx.x;
  const int n0 = blockIdx.y * 16;

  const float* wb = w + (size_t)b * LK;
  const float* V = value + (size_t)b * LK * DD + n0 + l16;

  v8f acc = {};
#pragma unroll 8
  for (int k0 = 0; k0 < LK; k0 += 4) {
    if ((k0 & 255) == 0 && k0 + 256 < LK)
      __builtin_prefetch(V + (size_t)(k0 + 256) * DD, 0, 0);
    const int kA = k0 + (half << 1);
    v2f a;
    a.x = wb[kA];
    a.y = wb[kA + 1];
    v2f bm;
    bm.x = V[(size_t)kA * DD];
    bm.y = V[(size_t)(kA + 1) * DD];
    acc = __builtin_amdgcn_wmma_f32_16x16x4_f32(
        /*neg_a=*/false, a, /*neg_b=*/false, bm,
        /*c_mod=*/(short)0, acc, /*reuse_a=*/false, /*reuse_b=*/false);
  }

  // C/D layout: VGPR i holds row M = i + 8*half, N = l16. All 64 query rows
  // of the output are identical -> broadcast across the 4 M tiles.
  float* ob = out + (size_t)b * LQ * DD + n0 + l16;
#pragma unroll
  for (int i = 0; i < 8; ++i) {
    const int row = i + (half << 3);
    const float d = acc[i];
#pragma unroll
    for (int mt = 0; mt < 4; ++mt) ob[(size_t)(mt * 16 + row) * DD] = d;
  }
}

// ---------------------------------------------------------------------------
// Inputs (setup_inputs order): query, key, value, valid_len, Wk, Wq, v
// query/Wk are mathematically dead: the per-query score component cancels in
// the softmax (see analysis); valid_len >= 1 guarantees correctness.
// ---------------------------------------------------------------------------
extern "C" void kernel_launch(void* const* d_in, const int* in_sizes, int n_in,
                              void* d_out, int out_size, void* d_ws,
                              size_t ws_size, hipStream_t stream) {
  const float* key = (const float*)d_in[1];
  const float* value = (const float*)d_in[2];
  const int* vlen = (const int*)d_in[3];
  const float* Wq = (const float*)d_in[5];
  const float* v = (const float*)d_in[6];
  float* out = (float*)d_out;

  float* ws_u2 = (float*)d_ws;     // 256 floats
  float* ws_w = ws_u2 + UU;        // BB*LK floats

  mlpattn_prep<<<1, 256, 0, stream>>>(Wq, v, ws_u2);
  mlpattn_softmax<<<BB, 256, 0, stream>>>(key, vlen, ws_u2, ws_w);
  mlpattn_av_wmma<<<dim3(BB, DD / 16), 32, 0, stream>>>(ws_w, value, out);

  (void)d_in;
  (void)in_sizes;
  (void)n_in;
  (void)out_size;
  (void)ws_size;
}